// MultiTaskGNN_v1_78795470012786
// MI455X (gfx1250) — compile-verified
//
#include <hip/hip_runtime.h>
#include <hip/hip_bf16.h>

// ---------------------------------------------------------------------------
// MultiTaskGNN (2x GINConv + mean-pool + MLP head) for MI455X / gfx1250.
// Memory/atomic bound problem; GEMMs done with exact-precision f32 WMMA
// (V_WMMA_F32_16X16X4_F32), which is the roofline-correct choice here.
// ---------------------------------------------------------------------------

typedef float v2f __attribute__((ext_vector_type(2)));
typedef float v8f __attribute__((ext_vector_type(8)));

#define HID 128

// ---------------------------------------------------------------------------
// 0) zero accumulator region (float4 grid-stride)
// ---------------------------------------------------------------------------
__global__ void zero_f4_kernel(float4* __restrict__ p, long n4) {
  long i = (long)blockIdx.x * blockDim.x + threadIdx.x;
  long stride = (long)gridDim.x * blockDim.x;
  float4 z; z.x = 0.f; z.y = 0.f; z.z = 0.f; z.w = 0.f;
  for (; i < n4; i += stride) p[i] = z;
}

// ---------------------------------------------------------------------------
// 1) edge scatter-add: agg[dst] += x[src], feature dim = 64
//    thread = edge*16 + quad ; float4 gather + 4 f32 global atomics
// ---------------------------------------------------------------------------
__global__ __launch_bounds__(256) void scatter_add64_kernel(
    const float* __restrict__ x, const long long* __restrict__ ei,
    float* __restrict__ agg, int n_edges) {
  int idx = blockIdx.x * blockDim.x + threadIdx.x;
  int e = idx >> 4;
  if (e >= n_edges) return;
  int q = idx & 15;
  long long src = ei[e];
  long long dst = ei[n_edges + e];
  const float4 v = *(const float4*)(x + src * 64 + q * 4);
  float* a = agg + dst * 64 + q * 4;
  atomicAdd(a + 0, v.x);
  atomicAdd(a + 1, v.y);
  atomicAdd(a + 2, v.z);
  atomicAdd(a + 3, v.w);
}

// feature dim = 128 version (thread = edge*32 + quad)
__global__ __launch_bounds__(256) void scatter_add128_kernel(
    const float* __restrict__ x, const long long* __restrict__ ei,
    float* __restrict__ agg, int n_edges) {
  int idx = blockIdx.x * blockDim.x + threadIdx.x;
  int e = idx >> 5;
  if (e >= n_edges) return;
  int q = idx & 31;
  long long src = ei[e];
  long long dst = ei[n_edges + e];
  const float4 v = *(const float4*)(x + src * 128 + q * 4);
  float* a = agg + dst * 128 + q * 4;
  atomicAdd(a + 0, v.x);
  atomicAdd(a + 1, v.y);
  atomicAdd(a + 2, v.z);
  atomicAdd(a + 3, v.w);
}

// ---------------------------------------------------------------------------
// 2) fused GIN MLP:  hout = relu( relu((x+agg) @ W1 + b1) @ W2 + b2 )
//    Block = 256 threads = 8 waves; each block owns a 16-node row tile.
//    Wave w computes output columns [16w, 16w+16) via f32 WMMA 16x16x4.
//    A-fragment layout (ISA 7.12.2, 32-bit A 16x4): lanes 0-15 hold K=k0,k0+1
//    for M=lane; lanes 16-31 hold K=k0+2,k0+3 for M=lane-16.
//    B/C/D: rows striped across lanes within a VGPR.
// ---------------------------------------------------------------------------
template <int KDIM>
__global__ __launch_bounds__(256) void gin_mlp_kernel(
    const float* __restrict__ xin, const float* __restrict__ agg,
    const float* __restrict__ W1, const float* __restrict__ b1,
    const float* __restrict__ W2, const float* __restrict__ b2,
    float* __restrict__ hout, int n_nodes) {
  __shared__ float T0[16 * KDIM];  // (x + agg) row tile
  __shared__ float T1[16 * HID];   // hidden activations after relu

  const int tid = threadIdx.x;
  const int row0 = blockIdx.x * 16;

  // stage (x + agg) tile into LDS (16*KDIM is a multiple of 256)
  for (int i = tid; i < 16 * KDIM; i += 256) {
    const int r = i / KDIM;
    const int c = i - r * KDIM;
    const int node = row0 + r;
    float v = 0.0f;
    if (node < n_nodes) {
      const long o = (long)node * KDIM + c;
      v = xin[o] + agg[o];
    }
    T0[i] = v;
  }
  __syncthreads();

  const int lane = tid & 31;
  const int wave = tid >> 5;      // N-tile index 0..7
  const int col = lane & 15;      // M (for A) / N (for B,C,D)
  const int hi = lane >> 4;       // K half-select
  const int n0 = wave * 16;

  // ---- GEMM 1: (16 x KDIM) @ (KDIM x 128) ----
  v8f acc = {0.f, 0.f, 0.f, 0.f, 0.f, 0.f, 0.f, 0.f};
#pragma unroll
  for (int k0 = 0; k0 < KDIM; k0 += 4) {
    const int ka = k0 + 2 * hi;
    v2f a, b;
    a.x = T0[col * KDIM + ka];
    a.y = T0[col * KDIM + ka + 1];
    b.x = W1[ka * HID + n0 + col];
    b.y = W1[(ka + 1) * HID + n0 + col];
    acc = __builtin_amdgcn_wmma_f32_16x16x4_f32(false, a, false, b,
                                                (short)0, acc, false, false);
  }
  {
    const float bias = b1[n0 + col];
#pragma unroll
    for (int r = 0; r < 8; ++r) {
      float v = acc[r] + bias;
      T1[(r + 8 * hi) * HID + n0 + col] = v > 0.f ? v : 0.f;
    }
  }
  __syncthreads();

  // ---- GEMM 2: (16 x 128) @ (128 x 128) ----
  v8f acc2 = {0.f, 0.f, 0.f, 0.f, 0.f, 0.f, 0.f, 0.f};
#pragma unroll
  for (int k0 = 0; k0 < HID; k0 += 4) {
    const int ka = k0 + 2 * hi;
    v2f a, b;
    a.x = T1[col * HID + ka];
    a.y = T1[col * HID + ka + 1];
    b.x = W2[ka * HID + n0 + col];
    b.y = W2[(ka + 1) * HID + n0 + col];
    acc2 = __builtin_amdgcn_wmma_f32_16x16x4_f32(false, a, false, b,
                                                 (short)0, acc2, false, false);
  }
  const float bias2 = b2[n0 + col];
#pragma unroll
  for (int r = 0; r < 8; ++r) {
    const int node = row0 + r + 8 * hi;
    if (node < n_nodes) {
      float v = acc2[r] + bias2;
      hout[(long)node * HID + n0 + col] = v > 0.f ? v : 0.f;
    }
  }
}

// ---------------------------------------------------------------------------
// 3) mean-pool scatter: pooled[batch[i]] += h2[i]; counts[batch[i]] += 1
// ---------------------------------------------------------------------------
__global__ __launch_bounds__(256) void pool_scatter_kernel(
    const float* __restrict__ h, const long long* __restrict__ batch,
    float* __restrict__ pooled, float* __restrict__ counts, int n_nodes) {
  int idx = blockIdx.x * blockDim.x + threadIdx.x;
  int node = idx >> 5;
  if (node >= n_nodes) return;
  int q = idx & 31;
  long long g = batch[node];
  const float4 v = *(const float4*)(h + (long)node * 128 + q * 4);
  float* p = pooled + g * 128 + q * 4;
  atomicAdd(p + 0, v.x);
  atomicAdd(p + 1, v.y);
  atomicAdd(p + 2, v.z);
  atomicAdd(p + 3, v.w);
  if (q == 0) atomicAdd(&counts[g], 1.0f);
}

// ---------------------------------------------------------------------------
// 4) head: out = relu(pooled/count @ Wl1 + bl1) @ Wl2 + bl2   (tiny: VALU)
//    one graph per 128-thread block
// ---------------------------------------------------------------------------
__global__ __launch_bounds__(128) void head_kernel(
    const float* __restrict__ pooled, const float* __restrict__ counts,
    const float* __restrict__ Wl1, const float* __restrict__ bl1,
    const float* __restrict__ Wl2, const float* __restrict__ bl2,
    float* __restrict__ out, int n_graphs) {
  __shared__ float row[128];
  __shared__ float hid[128];
  const int g = blockIdx.x;
  if (g >= n_graphs) return;
  const int t = threadIdx.x;
  const float cnt = counts[g];
  const float inv = 1.0f / (cnt > 1.0f ? cnt : 1.0f);
  row[t] = pooled[(long)g * 128 + t] * inv;
  __syncthreads();
  float acc = bl1[t];
#pragma unroll 8
  for (int k = 0; k < 128; ++k) acc += row[k] * Wl1[k * 128 + t];
  hid[t] = acc > 0.f ? acc : 0.f;
  __syncthreads();
  if (t < 3) {
    float o = bl2[t];
#pragma unroll 8
    for (int k = 0; k < 128; ++k) o += hid[k] * Wl2[k * 3 + t];
    out[(long)g * 3 + t] = o;
  }
}

// ---------------------------------------------------------------------------
// launch
// ---------------------------------------------------------------------------
extern "C" void kernel_launch(void* const* d_in, const int* in_sizes, int n_in,
                              void* d_out, int out_size, void* d_ws, size_t ws_size,
                              hipStream_t stream) {
  (void)n_in; (void)ws_size;

  const float* x = (const float*)d_in[0];
  const long long* ei = (const long long*)d_in[1];
  const long long* batch = (const long long*)d_in[2];
  const float* W1a = (const float*)d_in[3];
  const float* b1a = (const float*)d_in[4];
  const float* W2a = (const float*)d_in[5];
  const float* b2a = (const float*)d_in[6];
  const float* W1b = (const float*)d_in[7];
  const float* b1b = (const float*)d_in[8];
  const float* W2b = (const float*)d_in[9];
  const float* b2b = (const float*)d_in[10];
  const float* Wl1 = (const float*)d_in[11];
  const float* bl1 = (const float*)d_in[12];
  const float* Wl2 = (const float*)d_in[13];
  const float* bl2 = (const float*)d_in[14];
  float* out = (float*)d_out;

  const int n_nodes = in_sizes[0] / 64;      // 40000
  const int n_edges = in_sizes[1] / 2;       // 640000
  const int n_graphs = out_size / 3;         // 2000

  // workspace layout (floats). Accumulators first (zeroed each call).
  float* ws = (float*)d_ws;
  const long agg1_n = (long)n_nodes * 64;
  const long agg2_n = (long)n_nodes * 128;
  const long pooled_n = (long)n_graphs * 128;
  const long counts_n = 2048;  // padded
  float* agg1 = ws;
  float* agg2 = agg1 + agg1_n;
  float* pooled = agg2 + agg2_n;
  float* counts = pooled + pooled_n;
  float* h1 = counts + counts_n;
  float* h2 = h1 + (long)n_nodes * 128;

  // 0) zero accumulators
  {
    const long zero_floats = agg1_n + agg2_n + pooled_n + counts_n;
    const long n4 = zero_floats / 4;
    const int blocks = (int)((n4 + 255) / 256);
    zero_f4_kernel<<<blocks, 256, 0, stream>>>((float4*)ws, n4);
  }

  // 1) layer-A edge aggregation (dim 64)
  {
    const long threads = (long)n_edges * 16;
    scatter_add64_kernel<<<(int)((threads + 255) / 256), 256, 0, stream>>>(
        x, ei, agg1, n_edges);
  }

  // 2) layer-A MLP (WMMA)
  {
    const int blocks = (n_nodes + 15) / 16;
    gin_mlp_kernel<64><<<blocks, 256, 0, stream>>>(x, agg1, W1a, b1a, W2a, b2a,
                                                   h1, n_nodes);
  }

  // 3) layer-B edge aggregation (dim 128)
  {
    const long threads = (long)n_edges * 32;
    scatter_add128_kernel<<<(int)((threads + 255) / 256), 256, 0, stream>>>(
        h1, ei, agg2, n_edges);
  }

  // 4) layer-B MLP (WMMA)
  {
    const int blocks = (n_nodes + 15) / 16;
    gin_mlp_kernel<128><<<blocks, 256, 0, stream>>>(h1, agg2, W1b, b1b, W2b,
                                                    b2b, h2, n_nodes);
  }

  // 5) mean-pool scatter
  {
    const long threads = (long)n_nodes * 32;
    pool_scatter_kernel<<<(int)((threads + 255) / 256), 256, 0, stream>>>(
        h2, batch, pooled, counts, n_nodes);
  }

  // 6) head
  head_kernel<<<n_graphs, 128, 0, stream>>>(pooled, counts, Wl1, bl1, Wl2, bl2,
                                            out, n_graphs);
}